// GraphVAE_27367531610237
// MI455X (gfx1250) — compile-verified
//
#include <hip/hip_runtime.h>

// ---------------- problem constants (match reference) ----------------
#define N_NODES    50000
#define N_EDGES    800000
#define NUM_GRAPHS 256
#define FEAT       128
#define HID        256
#define LAT        64
#define MAXN       128
#define NBLK       ((N_NODES + 255) / 256)     // 196 scan blocks

typedef __attribute__((ext_vector_type(2))) float v2f;
typedef __attribute__((ext_vector_type(8))) float v8f;

// epilogue codes: 0 = raw, 1 = +bias, 2 = +bias,relu, 3 = +bias,sigmoid
__device__ __forceinline__ float epi_apply(float v, float b, int epi) {
    if (epi >= 1) v += b;
    if (epi == 2) v = fmaxf(v, 0.0f);
    if (epi == 3) v = 1.0f / (1.0f + __expf(-v));
    return v;
}

// ---------------------------------------------------------------------
// Wide fp32 WMMA GEMM: out[M,F] = A[M,K] @ B[K,F], B pre-packed in k-pairs
// (Bp[(k>>1)*2F + col*2 + (k&1)]).  One wave32 -> 16(M) x 128(F) tile,
// 8 accumulators, V_WMMA_F32_16X16X4_F32.  M%16==0, K%4==0, F%128==0.
// Per k-step: 1 A b64 + 8 B b64 loads feed 8 WMMAs.
// ---------------------------------------------------------------------
__global__ __launch_bounds__(256) void wmma_gemm_w128(
    const float* __restrict__ A, const float* __restrict__ Bp,
    const float* __restrict__ bias, float* __restrict__ out,
    int M, int K, int F, int epi)
{
    const int lane  = threadIdx.x & 31;
    const int numFt = F >> 7;
    const int numMt = M >> 4;
    const int wid   = blockIdx.x * 8 + (threadIdx.x >> 5);
    if (wid >= numFt * numMt) return;            // wave-uniform exit, EXEC stays all-1s
    const int ft = wid % numFt;
    const int mt = wid / numFt;
    const int m0 = mt << 4, f0 = ft << 7;
    const int lrow  = lane & 15;
    const int lhalf = lane >> 4;

    v8f zero = {};
    v8f acc[8];
#pragma unroll
    for (int t = 0; t < 8; ++t) acc[t] = zero;

    const float* arow = A + (size_t)(m0 + lrow) * K + (lhalf << 1);
    // lane's B base: pair row (kk>>1)+lhalf, column f0+lrow (+16t)
    const float* brow = Bp + (size_t)lhalf * (F << 1) + ((size_t)(f0 + lrow) << 1);

    for (int kk = 0; kk < K; kk += 4) {
        const v2f a = *(const v2f*)(arow + kk);                       // 8B aligned
        const float* bp = brow + (size_t)(kk >> 1) * (F << 1);
        v2f b[8];
#pragma unroll
        for (int t = 0; t < 8; ++t) b[t] = *(const v2f*)(bp + (t << 5));  // +16 cols = +32 floats
#pragma unroll
        for (int t = 0; t < 8; ++t)
            acc[t] = __builtin_amdgcn_wmma_f32_16x16x4_f32(false, a, false, b[t],
                                                           (short)0, acc[t], false, false);
    }

    const int c0 = f0 + lrow;
    float bb[8];
#pragma unroll
    for (int t = 0; t < 8; ++t) bb[t] = (epi >= 1) ? bias[c0 + 16 * t] : 0.0f;
#pragma unroll
    for (int r = 0; r < 8; ++r) {
        float* op = out + (size_t)(m0 + (lhalf << 3) + r) * F + c0;
#pragma unroll
        for (int t = 0; t < 8; ++t)
            op[16 * t] = epi_apply(acc[t][r], bb[t], epi);
    }
}

// Narrow 16x64 variant with direct (unpacked) B for F==64 GEMMs (mu/logvar).
__global__ __launch_bounds__(256) void wmma_gemm_w64(
    const float* __restrict__ A, const float* __restrict__ B,
    const float* __restrict__ bias, float* __restrict__ out,
    int M, int K, int F, int epi)
{
    const int lane  = threadIdx.x & 31;
    const int numFt = F >> 6;
    const int numMt = M >> 4;
    const int wid   = blockIdx.x * 8 + (threadIdx.x >> 5);
    if (wid >= numFt * numMt) return;
    const int ft = wid % numFt;
    const int mt = wid / numFt;
    const int m0 = mt << 4, f0 = ft << 6;
    const int lrow  = lane & 15;
    const int lhalf = lane >> 4;

    v8f acc0 = {}, acc1 = {}, acc2 = {}, acc3 = {};
    const float* arow = A + (size_t)(m0 + lrow) * K + (lhalf << 1);
    const float* bcol = B + f0 + lrow;
    for (int kk = 0; kk < K; kk += 4) {
        const v2f a = *(const v2f*)(arow + kk);
        const float* bp = bcol + (size_t)(kk + (lhalf << 1)) * F;
        v2f b0, b1, b2, b3;
        b0.x = bp[0];   b0.y = bp[F];
        b1.x = bp[16];  b1.y = bp[F + 16];
        b2.x = bp[32];  b2.y = bp[F + 32];
        b3.x = bp[48];  b3.y = bp[F + 48];
        acc0 = __builtin_amdgcn_wmma_f32_16x16x4_f32(false, a, false, b0, (short)0, acc0, false, false);
        acc1 = __builtin_amdgcn_wmma_f32_16x16x4_f32(false, a, false, b1, (short)0, acc1, false, false);
        acc2 = __builtin_amdgcn_wmma_f32_16x16x4_f32(false, a, false, b2, (short)0, acc2, false, false);
        acc3 = __builtin_amdgcn_wmma_f32_16x16x4_f32(false, a, false, b3, (short)0, acc3, false, false);
    }
    const int c0 = f0 + lrow;
    float bb0 = 0.f, bb1 = 0.f, bb2 = 0.f, bb3 = 0.f;
    if (epi >= 1) { bb0 = bias[c0]; bb1 = bias[c0 + 16]; bb2 = bias[c0 + 32]; bb3 = bias[c0 + 48]; }
#pragma unroll
    for (int r = 0; r < 8; ++r) {
        float* op = out + (size_t)(m0 + (lhalf << 3) + r) * F + c0;
        op[0]  = epi_apply(acc0[r], bb0, epi);
        op[16] = epi_apply(acc1[r], bb1, epi);
        op[32] = epi_apply(acc2[r], bb2, epi);
        op[48] = epi_apply(acc3[r], bb3, epi);
    }
}

// pack B[K,F] into k-pair-interleaved Bp so B fragments load as one b64
__global__ void pack_pairs(const float* __restrict__ B, float* __restrict__ Bp, int K, int F) {
    int idx = blockIdx.x * 256 + threadIdx.x;
    if (idx < K * F) {
        int k = idx / F, c = idx - k * F;
        Bp[((size_t)(k >> 1) * F + c) * 2 + (k & 1)] = B[idx];
    }
}

// ---------------- degree + CSR build ----------------
__global__ void zero_i32(int* p, int n) {
    int i = blockIdx.x * 256 + threadIdx.x;
    if (i < n) p[i] = 0;
}
__global__ void zero_f32(float* p, int n) {
    int i = blockIdx.x * 256 + threadIdx.x;
    if (i < n) p[i] = 0.0f;
}
__global__ void deg_accum(const int* __restrict__ dst, int* __restrict__ cnt, int E) {
    int e = blockIdx.x * 256 + threadIdx.x;
    if (e < E) atomicAdd(&cnt[dst[e]], 1);
}
__global__ void dinv_from_cnt(const int* __restrict__ cnt, float* __restrict__ d, int n) {
    int i = blockIdx.x * 256 + threadIdx.x;
    if (i < n) d[i] = rsqrtf((float)cnt[i] + 1.0f);
}
__global__ void block_sum(const int* __restrict__ v, int* __restrict__ bsum, int n) {
    __shared__ int sh[256];
    int i = blockIdx.x * 256 + threadIdx.x;
    sh[threadIdx.x] = (i < n) ? v[i] : 0;
    __syncthreads();
    for (int off = 128; off > 0; off >>= 1) {
        if (threadIdx.x < off) sh[threadIdx.x] += sh[threadIdx.x + off];
        __syncthreads();
    }
    if (threadIdx.x == 0) bsum[blockIdx.x] = sh[0];
}
__global__ void scan_bsums(int* bsum, int nb) {      // single-block exclusive scan (nb<=256)
    __shared__ int sh[256];
    int t = threadIdx.x;
    int orig = (t < nb) ? bsum[t] : 0;
    sh[t] = orig;
    __syncthreads();
    for (int off = 1; off < 256; off <<= 1) {
        int v = (t >= off) ? sh[t - off] : 0;
        __syncthreads();
        sh[t] += v;
        __syncthreads();
    }
    if (t < nb) bsum[t] = sh[t] - orig;              // exclusive
}
__global__ void block_scan(const int* __restrict__ v, const int* __restrict__ boff,
                           int* __restrict__ out, int n) {
    __shared__ int sh[256];
    int i = blockIdx.x * 256 + threadIdx.x;
    int orig = (i < n) ? v[i] : 0;
    sh[threadIdx.x] = orig;
    __syncthreads();
    for (int off = 1; off < 256; off <<= 1) {
        int val = (threadIdx.x >= off) ? sh[threadIdx.x - off] : 0;
        __syncthreads();
        sh[threadIdx.x] += val;
        __syncthreads();
    }
    if (i < n) out[i] = sh[threadIdx.x] - orig + boff[blockIdx.x];
}
__global__ void set_int(int* p, int v) { *p = v; }
__global__ void copy_i32(const int* __restrict__ a, int* __restrict__ b, int n) {
    int i = blockIdx.x * 256 + threadIdx.x;
    if (i < n) b[i] = a[i];
}
__global__ void csr_fill(const int* __restrict__ src, const int* __restrict__ dst,
                         int* __restrict__ cursor, int* __restrict__ col, int E) {
    int e = blockIdx.x * 256 + threadIdx.x;
    if (e < E) {
        int p = atomicAdd(&cursor[dst[e]], 1);
        col[p] = src[e];
    }
}

// ---------------------------------------------------------------------
// Fused GCN aggregate: wave per node, register accumulation (no atomics).
// act[n,:] = relu( sum_{s in N(n)} h[s,:]*dinv[s]*dinv[n] + h[n,:]*dinv[n]^2 + b )
// ---------------------------------------------------------------------
__global__ __launch_bounds__(256) void gcn_gather(
    const float* __restrict__ h, const float* __restrict__ dinv,
    const int* __restrict__ row_start, const int* __restrict__ col,
    const float* __restrict__ bias, float* __restrict__ act, int N)
{
    const int lane = threadIdx.x & 31;
    const int n = blockIdx.x * 8 + (threadIdx.x >> 5);
    if (n >= N) return;
    const float dn = dinv[n];
    const float* hn = h + (size_t)n * HID;
    float acc[8];
#pragma unroll
    for (int i = 0; i < 8; ++i)
        acc[i] = hn[i * 32 + lane] * (dn * dn) + bias[i * 32 + lane];

    const int beg = row_start[n], end = row_start[n + 1];
    for (int base = beg; base < end; base += 32) {
        const int m = end - base;
        const int cnt = m < 32 ? m : 32;
        int s = 0; float w = 0.0f;
        if (lane < cnt) { s = col[base + lane]; w = dinv[s]; }   // coalesced edge loads
        for (int j = 0; j < cnt; ++j) {
            const int   sj = __shfl(s, j, 32);
            const float wj = __shfl(w, j, 32) * dn;
            const float* hs = h + (size_t)sj * HID;
#pragma unroll
            for (int i = 0; i < 8; ++i)
                acc[i] = fmaf(hs[i * 32 + lane], wj, acc[i]);    // coalesced 128B reads
        }
    }
    float* an = act + (size_t)n * HID;
#pragma unroll
    for (int i = 0; i < 8; ++i)
        an[i * 32 + lane] = fmaxf(acc[i], 0.0f);
}

// one wave per node: g[batch[n],:] += h[n,:]  (12.8M atomics total, cheap)
__global__ __launch_bounds__(256) void pool_scatter(
    const int* __restrict__ batch, const float* __restrict__ h,
    float* __restrict__ g, int N)
{
    const int lane = threadIdx.x & 31;
    const int n = blockIdx.x * 8 + (threadIdx.x >> 5);
    if (n >= N) return;
    const int gidx = batch[n];
    const float* hp = h + (size_t)n * HID;
    float* gp = g + (size_t)gidx * HID;
#pragma unroll
    for (int i = 0; i < 8; ++i) {
        const int f = i * 32 + lane;
        atomicAdd(gp + f, hp[f]);
    }
}

// z = mu + eps * exp(0.5*logvar)
__global__ void reparam(const float* __restrict__ mu, const float* __restrict__ lv,
                        const float* __restrict__ eps, float* __restrict__ z, int n) {
    int i = blockIdx.x * 256 + threadIdx.x;
    if (i < n) z[i] = mu[i] + eps[i] * __expf(0.5f * lv[i]);
}

// ---------------------------------------------------------------------
extern "C" void kernel_launch(void* const* d_in, const int* in_sizes, int n_in,
                              void* d_out, int out_size, void* d_ws, size_t ws_size,
                              hipStream_t stream) {
    (void)in_sizes; (void)n_in; (void)out_size; (void)ws_size;
    const float* x     = (const float*)d_in[0];
    const int*   ei    = (const int*)d_in[1];
    const int*   batch = (const int*)d_in[2];
    const float* eps   = (const float*)d_in[3];
    const float* W1 = (const float*)d_in[4];  const float* b1 = (const float*)d_in[5];
    const float* W2 = (const float*)d_in[6];  const float* b2 = (const float*)d_in[7];
    const float* W3 = (const float*)d_in[8];  const float* b3 = (const float*)d_in[9];
    const float* W4 = (const float*)d_in[10]; const float* b4 = (const float*)d_in[11];
    const float* Wmu = (const float*)d_in[12]; const float* bmu = (const float*)d_in[13];
    const float* Wlv = (const float*)d_in[14]; const float* blv = (const float*)d_in[15];
    const float* Wd1 = (const float*)d_in[16]; const float* bd1 = (const float*)d_in[17];
    const float* Wadj = (const float*)d_in[18]; const float* badj = (const float*)d_in[19];
    const float* Wnode = (const float*)d_in[20]; const float* bnode = (const float*)d_in[21];

    float* out = (float*)d_out;

    // ---------------- workspace layout ----------------
    const size_t ACT = (size_t)N_NODES * HID;            // 12,800,000 floats
    float* dinv = (float*)d_ws;                          // 50,048
    float* bufA = dinv + 50048;                          // activations
    float* bufB = bufA + ACT;                            // h = act @ W
    float* Bp   = bufB + ACT;                            // packed weights (max 4,194,304)
    float* g    = Bp + (size_t)HID * MAXN * MAXN / 2;    // 256*16384/2 = 2,097,152 floats
    float* z    = g + NUM_GRAPHS * HID;
    float* dd   = z + NUM_GRAPHS * LAT;
    int* cnt       = (int*)(dd + NUM_GRAPHS * HID);
    int* row_start = cnt + 50048;                        // N_NODES+1 entries
    int* cursor    = row_start + 50176;
    int* col       = cursor + 50048;
    int* bsum      = col + N_EDGES;

    const int* srcI = ei;
    const int* dstI = ei + N_EDGES;

    // output layout: adj | node | mu | logvar
    float* adjO  = out;
    float* nodeO = out + (size_t)NUM_GRAPHS * MAXN * MAXN;
    float* muO   = nodeO + (size_t)NUM_GRAPHS * MAXN * FEAT;
    float* lvO   = muO + (size_t)NUM_GRAPHS * LAT;

    auto gemmW = [&](const float* A, const float* W, const float* bias, float* o,
                     int M, int K, int F, int epi) {
        pack_pairs<<<(K * F + 255) / 256, 256, 0, stream>>>(W, Bp, K, F);
        int tiles = (M / 16) * (F / 128);
        wmma_gemm_w128<<<(tiles + 7) / 8, 256, 0, stream>>>(A, Bp, bias, o, M, K, F, epi);
    };

    // degrees + dinv + CSR-by-dst
    zero_i32<<<NBLK, 256, 0, stream>>>(cnt, N_NODES);
    deg_accum<<<(N_EDGES + 255) / 256, 256, 0, stream>>>(dstI, cnt, N_EDGES);
    dinv_from_cnt<<<NBLK, 256, 0, stream>>>(cnt, dinv, N_NODES);
    block_sum<<<NBLK, 256, 0, stream>>>(cnt, bsum, N_NODES);
    scan_bsums<<<1, 256, 0, stream>>>(bsum, NBLK);
    block_scan<<<NBLK, 256, 0, stream>>>(cnt, bsum, row_start, N_NODES);
    set_int<<<1, 1, 0, stream>>>(row_start + N_NODES, N_EDGES);
    copy_i32<<<NBLK, 256, 0, stream>>>(row_start, cursor, N_NODES);
    csr_fill<<<(N_EDGES + 255) / 256, 256, 0, stream>>>(srcI, dstI, cursor, col, N_EDGES);

    // 4 GCN layers: GEMM -> fused gather(+self+bias+relu)
    const float* Ws[4] = {W1, W2, W3, W4};
    const float* bs[4] = {b1, b2, b3, b4};
    const float* act = x;
    int Kcur = FEAT;
    for (int l = 0; l < 4; ++l) {
        gemmW(act, Ws[l], nullptr, bufB, N_NODES, Kcur, HID, 0);
        gcn_gather<<<(N_NODES + 7) / 8, 256, 0, stream>>>(bufB, dinv, row_start, col, bs[l], bufA, N_NODES);
        act = bufA;
        Kcur = HID;
    }

    // global add pool
    zero_f32<<<(NUM_GRAPHS * HID + 255) / 256, 256, 0, stream>>>(g, NUM_GRAPHS * HID);
    pool_scatter<<<(N_NODES + 7) / 8, 256, 0, stream>>>(batch, bufA, g, N_NODES);

    // VAE head: mu/logvar straight into d_out (F=64 -> narrow kernel, direct B)
    wmma_gemm_w64<<<2, 256, 0, stream>>>(g, Wmu, bmu, muO, NUM_GRAPHS, HID, LAT, 1);
    wmma_gemm_w64<<<2, 256, 0, stream>>>(g, Wlv, blv, lvO, NUM_GRAPHS, HID, LAT, 1);
    reparam<<<(NUM_GRAPHS * LAT + 255) / 256, 256, 0, stream>>>(muO, lvO, eps, z, NUM_GRAPHS * LAT);

    // decoder
    gemmW(z, Wd1, bd1, dd, NUM_GRAPHS, LAT, 256, 2);                  // relu
    gemmW(dd, Wadj, badj, adjO, NUM_GRAPHS, HID, MAXN * MAXN, 3);     // sigmoid
    gemmW(dd, Wnode, bnode, nodeO, NUM_GRAPHS, HID, MAXN * FEAT, 1);  // bias
}